// AnchorPlusLoss_19524921327685
// MI455X (gfx1250) — compile-verified
//
#include <hip/hip_runtime.h>
#include <math.h>

typedef __attribute__((ext_vector_type(2))) float v2f;
typedef __attribute__((ext_vector_type(8))) float v8f;

#define B_ 4
#define N_ 2048
#define C_ 34
#define D_ 32   // feature dim (C_ - 2)

// Workspace layout (floats):
//   fn  : [B_*N_*D_]           normalized features
//   p   : [B_*N_*2]            2-D positions (emb[:,:2] + coords)
//   acc : [12]                 spatial[4], S[4] (softplus sums), cnt[4]
#define FN_OFF   0
#define P_OFF    (B_*N_*D_)
#define ACC_OFF  (P_OFF + B_*N_*2)

__global__ void init_acc_kernel(float* __restrict__ acc) {
    if (threadIdx.x < 12) acc[threadIdx.x] = 0.f;
}

__global__ void prep_kernel(const float* __restrict__ emb,
                            const float* __restrict__ coords,
                            float* __restrict__ fn,
                            float* __restrict__ p) {
    int idx = blockIdx.x * blockDim.x + threadIdx.x;   // point index in [0, B*N)
    if (idx >= B_ * N_) return;
    const float* e = emb + (size_t)idx * C_;
    p[idx * 2 + 0] = e[0] + coords[idx * 2 + 0];
    p[idx * 2 + 1] = e[1] + coords[idx * 2 + 1];
    float f[D_];
    float nrm = 0.f;
    #pragma unroll
    for (int k = 0; k < D_; ++k) { f[k] = e[2 + k]; nrm += f[k] * f[k]; }
    float inv = 1.f / fmaxf(sqrtf(nrm), 1e-8f);
    #pragma unroll
    for (int k = 0; k < D_; ++k) fn[(size_t)idx * D_ + k] = f[k] * inv;
}

__device__ __forceinline__ float softplusf(float x) {
    return x > 20.f ? x : log1pf(__expf(x));
}

// One wave per 16-row i-tile; 8 waves per block. Tiles = B * (N/16) = 512.
__global__ void tile_kernel(const float* __restrict__ fn,
                            const float* __restrict__ p,
                            float* __restrict__ acc) {
    const int wave   = threadIdx.x >> 5;
    const int lane   = threadIdx.x & 31;
    const int tile   = blockIdx.x * (blockDim.x >> 5) + wave;  // 0..511
    const int b      = tile >> 7;              // N_/16 = 128 tiles per batch
    const int ibase  = (tile & 127) << 4;
    const int nloc   = lane & 15;              // A row / B col / D col owned by lane
    const int koff   = (lane >> 4) << 1;       // K sub-slice: 0 or 2
    const int rowoff = (lane >> 4) << 3;       // D rows: 0..7 or 8..15

    const float* fnb = fn + (size_t)b * N_ * D_;
    const float* pb  = p  + (size_t)b * N_ * 2;

    // Preload A operand (i rows, all 8 K-slices of 4)
    v2f A[8];
    {
        const float* ar = fnb + (size_t)(ibase + nloc) * D_;
        #pragma unroll
        for (int k = 0; k < 8; ++k) {
            A[k].x = ar[4 * k + koff];
            A[k].y = ar[4 * k + koff + 1];
        }
    }
    // Positions of this lane's 8 output rows
    float pix[8], piy[8];
    #pragma unroll
    for (int v = 0; v < 8; ++v) {
        int r = ibase + rowoff + v;
        pix[v] = pb[r * 2 + 0];
        piy[v] = pb[r * 2 + 1];
    }

    float s[8];                 // per-row running sum of exp(x) (LSE without max: safe, x <= 35.2)
    #pragma unroll
    for (int v = 0; v < 8; ++v) s[v] = 0.f;
    float spatial = 0.f, cnt = 0.f;

    for (int jt = 0; jt < (N_ / 16); ++jt) {
        const int jbase = jt << 4;
        // B operand (j rows, transposed role)
        v2f Bv[8];
        const float* br = fnb + (size_t)(jbase + nloc) * D_;
        #pragma unroll
        for (int k = 0; k < 8; ++k) {
            Bv[k].x = br[4 * k + koff];
            Bv[k].y = br[4 * k + koff + 1];
        }
        // 16x16 Gram tile in full fp32: 8 chained K=4 WMMAs cover the 32-dim dot
        v8f Cm = {};
        #pragma unroll
        for (int k = 0; k < 8; ++k)
            Cm = __builtin_amdgcn_wmma_f32_16x16x4_f32(false, A[k], false, Bv[k],
                                                       (short)0, Cm, false, false);

        const float pjx = pb[(jbase + nloc) * 2 + 0];
        const float pjy = pb[(jbase + nloc) * 2 + 1];
        const int   gj  = jbase + nloc;
        #pragma unroll
        for (int v = 0; v < 8; ++v) {
            int gi = ibase + rowoff + v;
            float dx = pix[v] - pjx, dy = piy[v] - pjy;
            float sq = dx * dx + dy * dy;
            float dist = sq > 0.f ? sqrtf(sq) : 0.f;
            spatial += 1.f / (1.f + __expf(1.f - dist));     // sigmoid(dist - 1)
            bool pos = dist < 1.f;
            cnt += pos ? 1.f : 0.f;
            float x = (pos && gi != gj) ? (-32.f * (Cm[v] - 0.1f)) : 0.f;
            s[v] += __expf(x);                               // exp(0)=1 for masked entries, per ref
        }
    }

    // Row-wise LSE reduction: the 16 lanes sharing a row differ only in bits 0..3
    #pragma unroll
    for (int mask = 1; mask < 16; mask <<= 1) {
        #pragma unroll
        for (int v = 0; v < 8; ++v) s[v] += __shfl_xor(s[v], mask, 32);
    }
    if (nloc == 0) {
        float a = 0.f;
        #pragma unroll
        for (int v = 0; v < 8; ++v) a += softplusf(__logf(s[v]));
        atomicAdd(&acc[4 + b], a);      // S[b]: sum_i softplus(lse_i)
    }
    // Full-wave reduction of spatial sum and positive count
    #pragma unroll
    for (int mask = 1; mask < 32; mask <<= 1) {
        spatial += __shfl_xor(spatial, mask, 32);
        cnt     += __shfl_xor(cnt, mask, 32);
    }
    if (lane == 0) {
        atomicAdd(&acc[0 + b], spatial);
        atomicAdd(&acc[8 + b], cnt);
    }
}

__global__ void finalize_kernel(const float* __restrict__ acc, float* __restrict__ out) {
    float total = 0.f;
    #pragma unroll
    for (int b = 0; b < B_; ++b) {
        total += acc[b];                       // spatial loss
        float S = acc[4 + b];
        float c = acc[8 + b];                  // #(dist < 1), incl. diagonal
        float num_pos = c - (float)N_;         // eye removed from positive mask
        float num_neg = (float)N_ * (float)N_ - c;
        total += S / (num_pos + 1e-8f) + S / (num_neg + 1e-8f);
    }
    out[0] = total;
}

extern "C" void kernel_launch(void* const* d_in, const int* in_sizes, int n_in,
                              void* d_out, int out_size, void* d_ws, size_t ws_size,
                              hipStream_t stream) {
    const float* emb    = (const float*)d_in[0];   // [B,N,34] f32
    const float* coords = (const float*)d_in[1];   // [B,N,2]  f32
    float* ws  = (float*)d_ws;
    float* fn  = ws + FN_OFF;
    float* p   = ws + P_OFF;
    float* acc = ws + ACC_OFF;

    init_acc_kernel<<<1, 32, 0, stream>>>(acc);

    int npts = B_ * N_;
    prep_kernel<<<(npts + 255) / 256, 256, 0, stream>>>(emb, coords, fn, p);

    // 512 i-tiles, 8 waves (tiles) per 256-thread block -> 64 blocks
    tile_kernel<<<64, 256, 0, stream>>>(fn, p, acc);

    finalize_kernel<<<1, 1, 0, stream>>>(acc, (float*)d_out);
}